// UHBR_76519137345914
// MI455X (gfx1250) — compile-verified
//
#include <hip/hip_runtime.h>
#include <stddef.h>

#define NU   8039
#define NI   32770
#define NB   4771
#define NTOT (NU + NI + NB)      // 45580
#define EMB  32
#define NNZE 2000000
#define BATCHN 2048
#define KB   100
#define EPSF 1e-8f
#define L2NORM 1e-5f

typedef __attribute__((ext_vector_type(2))) float v2f;
typedef __attribute__((ext_vector_type(8))) float v8f;

__device__ __forceinline__ void atomAddF(float* p, float v) {
    (void)__hip_atomic_fetch_add(p, v, __ATOMIC_RELAXED, __HIP_MEMORY_SCOPE_AGENT);
}

__device__ __forceinline__ float feat_at(const float* uf, const float* itf,
                                         const float* bfv, int i, int d) {
    if (i < NU)      return uf[(size_t)i * EMB + d];
    if (i < NU + NI) return itf[(size_t)(i - NU) * EMB + d];
    return bfv[(size_t)(i - NU - NI) * EMB + d];
}

// ---------- zero workspace regions (grid-stride) ----------
__global__ void zero_k(float* p, long n) {
    long i = (long)blockIdx.x * blockDim.x + threadIdx.x;
    long stride = (long)gridDim.x * blockDim.x;
    for (; i < n; i += stride) p[i] = 0.0f;
}

// ---------- degree counts (integer-valued float atomics: exact) ----------
__global__ void degree_k(const int* __restrict__ rows, const int* __restrict__ cols,
                         float* dv, float* de) {
    int e = blockIdx.x * blockDim.x + threadIdx.x;
    if (e >= NNZE) return;
    atomAddF(dv + rows[e], 1.0f);
    atomAddF(de + cols[e], 1.0f);
}

// ---------- d -> 1/(sqrt(d)+eps), in place over dv|de (2N contiguous) ----------
__global__ void rsq_k(float* p, long n) {
    long i = (long)blockIdx.x * blockDim.x + threadIdx.x;
    if (i < n) p[i] = 1.0f / (sqrtf(p[i]) + EPSF);
}

// ---------- z = x * dvi ----------
__global__ void buildz_k(const float* uf, const float* itf, const float* bfv,
                         const float* __restrict__ dvi, float* __restrict__ z) {
    long t = (long)blockIdx.x * blockDim.x + threadIdx.x;
    if (t >= (long)NTOT * EMB) return;
    int i = (int)(t >> 5), d = (int)(t & 31);
    z[t] = feat_at(uf, itf, bfv, i, d) * dvi[i];
}

// ---------- edge scatter pass: dst[didx[e]] += src[sidx[e]] ----------
// one wave32 per edge, lane = emb dim: 128B coalesced load + 128B fp32 atomic.
// Everything (features + indices) is resident in the 192MB L2, so this runs
// at L2/atomic-unit bandwidth, not HBM.
__global__ void scatter_k(const int* __restrict__ sidx, const int* __restrict__ didx,
                          const float* __restrict__ src, float* __restrict__ dst) {
    long t = (long)blockIdx.x * blockDim.x + threadIdx.x;
    int e = (int)(t >> 5);
    int lane = (int)(t & 31);
    if (e >= NNZE) return;
    int s = sidx[e], d = didx[e];
    float val = src[(size_t)s * EMB + lane];
    atomAddF(dst + (size_t)d * EMB + lane, val);
}

// ---------- row scale: arr[i,:] *= s[i] ----------
__global__ void rowscale_k(float* __restrict__ arr, const float* __restrict__ s) {
    long t = (long)blockIdx.x * blockDim.x + threadIdx.x;
    if (t >= (long)NTOT * EMB) return;
    arr[t] *= s[t >> 5];
}

// ---------- all_embeds = x/2 + (v*dvi)/3 ; block-reduce sum of squares ----------
__global__ void final_k(const float* uf, const float* itf, const float* bfv,
                        const float* __restrict__ dvi, const float* __restrict__ v,
                        float* __restrict__ all, float* __restrict__ partials) {
    __shared__ float sdata[256];
    long t = (long)blockIdx.x * blockDim.x + threadIdx.x;
    float sq = 0.0f;
    if (t < (long)NTOT * EMB) {
        int i = (int)(t >> 5), d = (int)(t & 31);
        float val = feat_at(uf, itf, bfv, i, d) * 0.5f + v[t] * dvi[i] * (1.0f / 3.0f);
        all[t] = val;
        sq = val * val;
    }
    sdata[threadIdx.x] = sq;
    __syncthreads();
    for (int s = 128; s > 0; s >>= 1) {
        if (threadIdx.x < (unsigned)s) sdata[threadIdx.x] += sdata[threadIdx.x + s];
        __syncthreads();
    }
    if (threadIdx.x == 0) partials[blockIdx.x] = sdata[0];
}

// ---------- deterministic final loss reduce ----------
__global__ void loss_k(const float* __restrict__ partials, int nb, float* __restrict__ out_loss) {
    __shared__ float sdata[256];
    float s = 0.0f;
    for (int i = threadIdx.x; i < nb; i += 256) s += partials[i];
    sdata[threadIdx.x] = s;
    __syncthreads();
    for (int k = 128; k > 0; k >>= 1) {
        if (threadIdx.x < (unsigned)k) sdata[threadIdx.x] += sdata[threadIdx.x + k];
        __syncthreads();
    }
    if (threadIdx.x == 0) *out_loss = L2NORM * sdata[0];
}

// ---------- pred + user_score_bound via V_WMMA_F32_16X16X4_F32 ----------
// one wave per (batch b, tile kt of 16 bundles). A = 16 bundle rows x 4 emb,
// B = ue broadcast across N columns, 8 K-steps of 4. user_bound folded in as
// row M=4 of the last tile (kt==6). Invalid rows load from a zeroed 32-float
// workspace row so every lane does an unconditional aligned b64 load and EXEC
// stays all-ones (required by WMMA; also kills the exec-mask branch overhead).
// D column N=0 read from lanes 0 and 16.
__global__ void pred_k(const float* __restrict__ all, const int* __restrict__ users,
                       const int* __restrict__ bundles, const float* __restrict__ ubound,
                       const float* __restrict__ zrow,
                       float* __restrict__ pred, float* __restrict__ usb) {
    int wave = (blockIdx.x * blockDim.x + threadIdx.x) >> 5;
    int lane = threadIdx.x & 31;
    int b = wave / 7;
    int kt = wave % 7;

    const float* bfbase = all + (size_t)(NU + NI) * EMB;
    const float* ue = all + (size_t)users[b] * EMB;

    int m = lane & 15;      // A-matrix row handled by this lane
    int khalf = lane >> 4;  // K half (0 -> K=0,1 ; 1 -> K=2,3) per f32 A/B layout
    int kk = kt * 16 + m;

    const float* arow;
    if (kk < KB)                 arow = bfbase + (size_t)bundles[b * KB + kk] * EMB;
    else if (kt == 6 && m == 4)  arow = ubound;   // usb row
    else                         arow = zrow;     // zero padding row

    v8f c = {};
#pragma unroll
    for (int s = 0; s < 8; ++s) {
        int e0 = s * 4 + khalf * 2;               // even -> 8B aligned
        v2f a  = *(const v2f*)(arow + e0);
        v2f bb = *(const v2f*)(ue + e0);
        c = __builtin_amdgcn_wmma_f32_16x16x4_f32(false, a, false, bb,
                                                  (short)0, c, false, false);
    }

    // D layout: VGPR j, lanes 0-15: (M=j, N=lane); lanes 16-31: (M=8+j, N=lane-16)
    if (lane == 0) {
#pragma unroll
        for (int j = 0; j < 8; ++j) {
            int kkk = kt * 16 + j;
            if (kkk < KB)                 pred[(size_t)b * KB + kkk] = c[j];
            else if (kt == 6 && j == 4)   usb[b] = c[j];
        }
    } else if (lane == 16) {
#pragma unroll
        for (int j = 0; j < 8; ++j) {
            int kkk = kt * 16 + 8 + j;
            if (kkk < KB) pred[(size_t)b * KB + kkk] = c[j];
        }
    }
}

extern "C" void kernel_launch(void* const* d_in, const int* in_sizes, int n_in,
                              void* d_out, int out_size, void* d_ws, size_t ws_size,
                              hipStream_t stream) {
    const float* uf      = (const float*)d_in[0];
    const float* itf     = (const float*)d_in[1];
    const float* bfv     = (const float*)d_in[2];
    const float* ubound  = (const float*)d_in[3];
    const int*   rows    = (const int*)d_in[4];
    const int*   cols    = (const int*)d_in[5];
    const int*   users   = (const int*)d_in[6];
    const int*   bundles = (const int*)d_in[7];
    float* out = (float*)d_out;

    float* ws   = (float*)d_ws;
    float* dvde = ws;                         // dv: [0,N), de: [N,2N)
    float* z    = dvde + 2L * NTOT;           // reused as all_embeds later
    float* u    = z + 32L * NTOT;
    float* v    = u + 32L * NTOT;
    float* partials = v + 32L * NTOT;
    float* zrow = partials + 8192;            // 32-float zero row for WMMA padding

    const long elems = (long)NTOT * EMB;      // 1,458,560
    const int  eb    = (int)((elems + 255) / 256);          // 5698 blocks
    const int  edgeb = (NNZE * 32 + 255) / 256;             // 250,000 blocks

    // zero dv|de, u|v (contiguous pairs), and the padding row
    zero_k<<<2048, 256, 0, stream>>>(dvde, 2L * NTOT);
    zero_k<<<4096, 256, 0, stream>>>(u, 64L * NTOT);
    zero_k<<<1, 32, 0, stream>>>(zrow, 32L);

    degree_k<<<(NNZE + 255) / 256, 256, 0, stream>>>(rows, cols, dvde, dvde + NTOT);
    rsq_k<<<(int)((2L * NTOT + 255) / 256), 256, 0, stream>>>(dvde, 2L * NTOT);

    buildz_k<<<eb, 256, 0, stream>>>(uf, itf, bfv, dvde, z);
    scatter_k<<<edgeb, 256, 0, stream>>>(rows, cols, z, u);       // u[c] += z[r]
    rowscale_k<<<eb, 256, 0, stream>>>(u, dvde + NTOT);           // u *= dei
    scatter_k<<<edgeb, 256, 0, stream>>>(cols, rows, u, v);       // v[r] += u[c]

    final_k<<<eb, 256, 0, stream>>>(uf, itf, bfv, dvde, v, z, partials);
    loss_k<<<1, 256, 0, stream>>>(partials, eb, out + BATCHN * KB + BATCHN);

    // 2048 batches x 7 tiles = 14336 waves; 8 waves/block -> 1792 blocks exact
    pred_k<<<1792, 256, 0, stream>>>(z, users, bundles, ubound, zrow,
                                     out, out + BATCHN * KB);
}